// MultiPartLoss_31001073942778
// MI455X (gfx1250) — compile-verified
//
#include <hip/hip_runtime.h>

typedef float v2f  __attribute__((ext_vector_type(2)));
typedef float v8f  __attribute__((ext_vector_type(8)));

#define SGRID 7
#define NCLS 20
#define DPC 30                       // floats per cell
#define CPB 256                      // cells per block
#define FPB (CPB * DPC)              // 7680 floats per block per tensor
#define VPB (FPB / 4)                // 1920 16-byte vectors per block per tensor
#define L_COORD 5.0f
#define L_NOOBJ 0.5f
#define GRID_WH 64.0f                // 448/7
#define IMG_WH 448.0f

// CDNA5 async direct global->LDS 128-bit load (bypasses VGPRs, ASYNCcnt-tracked).
// lds_off = workgroup-relative LDS byte offset (low 32 bits of the flat address).
__device__ __forceinline__ void async_load_b128(unsigned lds_off, const float* gptr) {
    asm volatile("global_load_async_to_lds_b128 %0, %1, off th:TH_LOAD_NT"
                 :: "v"(lds_off), "v"(gptr) : "memory");
}
__device__ __forceinline__ void wait_async0() {
    asm volatile("s_wait_asynccnt 0x0" ::: "memory");
}

// Full-wave32 sum using V_WMMA_F32_16X16X4_F32:
//   A (16x4): lane L holds {v_L, 0} -> D[m][n] = v_m + v_{m+16} (B = ones 4x16)
//   D VGPR r: lanes 0-15 = row r, lanes 16-31 = row 8+r
//   sum of 8 D regs: lanes 0-15 hold sum{0..7,16..23}, lanes 16-31 hold the rest;
//   one ds_swizzle SWAPX16 (lane^16) finishes the 32-lane total on every lane.
__device__ __forceinline__ float wave_reduce_wmma(float v) {
    v2f a; a.x = v;    a.y = 0.0f;
    v2f b; b.x = 1.0f; b.y = 1.0f;
    v8f c = {};
    c = __builtin_amdgcn_wmma_f32_16x16x4_f32(false, a, false, b,
                                              (short)0, c, false, false);
    float s8 = c[0] + c[1] + c[2] + c[3] + c[4] + c[5] + c[6] + c[7];
    int o = __builtin_amdgcn_ds_swizzle(__float_as_int(s8), 0x401F); // SWAPX16
    return s8 + __int_as_float(o);
}

__device__ __forceinline__ float iou_box(float px, float py, float pw, float ph,
                                         float tx, float ty, float tw, float th) {
    float xA = fmaxf(px - 0.5f * pw, tx - 0.5f * tw);
    float yA = fmaxf(py - 0.5f * ph, ty - 0.5f * th);
    float xB = fminf(px + 0.5f * pw, tx + 0.5f * tw);
    float yB = fminf(py + 0.5f * ph, ty + 0.5f * th);
    float inter = fmaxf(0.0f, xB - xA + 1.0f) * fmaxf(0.0f, yB - yA + 1.0f);
    return inter / (pw * ph + tw * th - inter);
}

__global__ __launch_bounds__(256) void yolo_loss_pass1(
        const float* __restrict__ preds, const float* __restrict__ targets,
        float* __restrict__ partials, long long total_floats) {
    __shared__ __align__(16) float lp[FPB];
    __shared__ __align__(16) float lt[FPB];
    __shared__ float wsum[8];

    const int tid = threadIdx.x;
    const long long base = (long long)blockIdx.x * FPB;
    const long long remain = total_floats - base;

    if (remain >= (long long)FPB) {
        // async global->LDS streaming copy, fully coalesced 128-bit, NT-hinted
        const unsigned lp_off = (unsigned)(size_t)(void*)lp;
        const unsigned lt_off = (unsigned)(size_t)(void*)lt;
        const float* gp = preds + base;
        const float* gt = targets + base;
        #pragma unroll
        for (int i = 0; i < (VPB + 255) / 256; ++i) {
            int idx = tid + i * 256;
            if (idx < VPB) {
                async_load_b128(lp_off + (unsigned)idx * 16u, gp + idx * 4);
                async_load_b128(lt_off + (unsigned)idx * 16u, gt + idx * 4);
            }
        }
        wait_async0();
    } else {
        for (long long i = tid; i < remain; i += 256) {
            lp[i] = preds[base + i];
            lt[i] = targets[base + i];
        }
    }
    __syncthreads();

    const long long total_cells = total_floats / DPC;
    const long long cell = (long long)blockIdx.x * CPB + tid;
    float ls = 0.0f;
    if (cell < total_cells) {
        const float* p = lp + tid * DPC;
        const float* t = lt + tid * DPC;
        const int sidx = (int)(cell % (SGRID * SGRID));
        const float col = (float)(sidx % SGRID);
        const float row = (float)(sidx / SGRID);

        // no-object confidence term over ALL cells
        float dc0 = p[NCLS + 0] - t[NCLS + 0];
        float dc1 = p[NCLS + 1] - t[NCLS + 1];
        ls = L_NOOBJ * (dc0 * dc0 + dc1 * dc1);

        // decode + IoU for both boxes (box b of preds vs box b of targets)
        float iou0, iou1;
        {
            float px = (p[22] + col) * GRID_WH, py = (p[23] + row) * GRID_WH;
            float pw = p[24] * IMG_WH,          ph = p[25] * IMG_WH;
            float tx = (t[22] + col) * GRID_WH, ty = (t[23] + row) * GRID_WH;
            float tw = t[24] * IMG_WH,          th = t[25] * IMG_WH;
            iou0 = iou_box(px, py, pw, ph, tx, ty, tw, th);
        }
        {
            float px = (p[26] + col) * GRID_WH, py = (p[27] + row) * GRID_WH;
            float pw = p[28] * IMG_WH,          ph = p[29] * IMG_WH;
            float tx = (t[26] + col) * GRID_WH, ty = (t[27] + row) * GRID_WH;
            float tw = t[28] * IMG_WH,          th = t[29] * IMG_WH;
            iou1 = iou_box(px, py, pw, ph, tx, ty, tw, th);
        }
        // argmax picks first max on tie -> best=1 only when strictly greater
        const int best = (iou1 > iou0) ? 1 : 0;
        const int bo = 22 + 4 * best;
        const int co = NCLS + best;

        // objectness = (sum of target class probs > 0)
        float tsum = 0.0f;
        #pragma unroll
        for (int k = 0; k < NCLS; ++k) tsum += t[k];
        const float obj = (tsum > 0.0f) ? 1.0f : 0.0f;

        // class-prob SSE
        float cls = 0.0f;
        #pragma unroll
        for (int k = 0; k < NCLS; ++k) { float d = p[k] - t[k]; cls += d * d; }

        float dcb = p[co] - t[co];
        float dx  = p[bo + 0] - t[bo + 0];
        float dy  = p[bo + 1] - t[bo + 1];
        float dsw = sqrtf(p[bo + 2]) - sqrtf(t[bo + 2]);
        float dsh = sqrtf(p[bo + 3]) - sqrtf(t[bo + 3]);

        ls += obj * ((1.0f - L_NOOBJ) * dcb * dcb
                     + cls
                     + L_COORD * (dx * dx + dy * dy)
                     + L_COORD * (dsw * dsw + dsh * dsh));
    }

    // wave32 reduction via WMMA (EXEC all-ones here: all lanes converged)
    const float w = wave_reduce_wmma(ls);
    const int lane = tid & 31, warp = tid >> 5;
    if (lane == 0) wsum[warp] = w;
    __syncthreads();
    if (tid == 0) {
        float s = 0.0f;
        #pragma unroll
        for (int i = 0; i < 8; ++i) s += wsum[i];
        partials[blockIdx.x] = s;
    }
}

__global__ __launch_bounds__(256) void yolo_loss_pass2(
        const float* __restrict__ partials, int nparts, float scale,
        float* __restrict__ out) {
    __shared__ float wsum[8];
    const int tid = threadIdx.x;
    float s = 0.0f;
    for (int i = tid; i < nparts; i += 256) s += partials[i];
    const float w = wave_reduce_wmma(s);
    if ((tid & 31) == 0) wsum[tid >> 5] = w;
    __syncthreads();
    if (tid == 0) {
        float tot = 0.0f;
        #pragma unroll
        for (int i = 0; i < 8; ++i) tot += wsum[i];
        out[0] = tot * scale;
    }
}

extern "C" void kernel_launch(void* const* d_in, const int* in_sizes, int n_in,
                              void* d_out, int out_size, void* d_ws, size_t ws_size,
                              hipStream_t stream) {
    const float* preds   = (const float*)d_in[0];
    const float* targets = (const float*)d_in[1];
    const long long total_floats = (long long)in_sizes[0];
    const long long cells = total_floats / DPC;           // 32768*49
    const int nblocks = (int)((cells + CPB - 1) / CPB);   // 6272
    const long long N = cells / (SGRID * SGRID);          // 32768
    const float scale = (float)(1.0 / (double)N);

    float* partials = (float*)d_ws;
    yolo_loss_pass1<<<nblocks, 256, 0, stream>>>(preds, targets, partials, total_floats);
    yolo_loss_pass2<<<1, 256, 0, stream>>>(partials, nblocks, scale, (float*)d_out);
}